// Attention_13907104104475
// MI455X (gfx1250) — compile-verified
//
#include <hip/hip_runtime.h>
#include <hip/hip_bf16.h>

// ---------------------------------------------------------------------------
// Attention forward for MI455X (gfx1250), f32 WMMA 16x16x4 path.
//   VOCAB=50257, SEQ=2048, EMB=512, BATCH=4
// Each GEMM wave computes a 32x32 tile (2x2 register blocking, 4 v8f accs)
// to halve vector-memory instructions per v_wmma and break RAW chains.
// ---------------------------------------------------------------------------

#define VOCAB 50257
#define SEQ   2048
#define EMB   512
#define BATCH 4

typedef __attribute__((ext_vector_type(2))) float v2f;
typedef __attribute__((ext_vector_type(8))) float v8f;

#define WMMA_F32(acc, a, b)                                                    \
    acc = __builtin_amdgcn_wmma_f32_16x16x4_f32(false, (a), false, (b),        \
                                                (short)0, (acc), false, false)

// ---------------------------------------------------------------------------
// 1) Embedding gather + positional add:
//    out[b,s,e] = W_emb[e*VOCAB + tok[b,s]] + W_pos[s,e]
// ---------------------------------------------------------------------------
__global__ void embed_kernel(const int* __restrict__ tok,      // (B,S)
                             const float* __restrict__ Wemb,   // (E,V)
                             const float* __restrict__ Wpos,   // (S,E)
                             float* __restrict__ out)          // (B,S,E)
{
    const int bs = blockIdx.x;          // 0 .. B*S-1
    const int s  = bs & (SEQ - 1);
    const int t  = tok[bs];
    const float* pos = Wpos + (size_t)s * EMB;
    float* o = out + (size_t)bs * EMB;
    for (int e = threadIdx.x; e < EMB; e += blockDim.x)
        o[e] = Wemb[(size_t)e * VOCAB + t] + pos[e];
}

// ---------------------------------------------------------------------------
// 2) Projection GEMM: C[b,m,n] = sum_k A[b,m,k] * W[n,k] + bias[n]
//    A: (B,SEQ,EMB) row-major; W: (EMB,EMB) row-major. Wave = 32x32 tile.
//    f32 WMMA 16x16x4 fragment layout:
//      A/B frag: lane<16 -> row/col=lane, K=k+{0,1}; lane>=16 -> K=k+{2,3}
//      D VGPR r: M = r + 8*(lane>=16), N = lane&15
// ---------------------------------------------------------------------------
__global__ void proj_gemm(const float* __restrict__ A,
                          const float* __restrict__ W,
                          const float* __restrict__ bias,
                          float* __restrict__ C)
{
    const int n0   = blockIdx.x * 32;
    const int m0   = blockIdx.y * 32;
    const int b    = blockIdx.z;
    const int lane = threadIdx.x;
    const int half = lane >> 4;         // 0 or 1
    const int l16  = lane & 15;

    const float* a0p = A + ((size_t)b * SEQ + m0 + l16) * EMB + 2 * half;
    const float* a1p = a0p + (size_t)16 * EMB;
    const float* b0p = W + (size_t)(n0 + l16) * EMB + 2 * half;
    const float* b1p = b0p + (size_t)16 * EMB;

    v8f acc00 = {}, acc01 = {}, acc10 = {}, acc11 = {};
#pragma unroll 4
    for (int k = 0; k < EMB; k += 4) {
        const v2f a0 = *(const v2f*)(a0p + k);
        const v2f a1 = *(const v2f*)(a1p + k);
        const v2f w0 = *(const v2f*)(b0p + k);
        const v2f w1 = *(const v2f*)(b1p + k);
        WMMA_F32(acc00, a0, w0);
        WMMA_F32(acc01, a0, w1);
        WMMA_F32(acc10, a1, w0);
        WMMA_F32(acc11, a1, w1);
    }

    const float bi0 = bias[n0 + l16];
    const float bi1 = bias[n0 + 16 + l16];
    float* cp = C + ((size_t)b * SEQ + m0) * EMB + n0;
#pragma unroll
    for (int r = 0; r < 8; ++r) {
        const int m = r + 8 * half;
        cp[(size_t)m * EMB + l16]             = acc00[r] + bi0;
        cp[(size_t)m * EMB + 16 + l16]        = acc01[r] + bi1;
        cp[(size_t)(m + 16) * EMB + l16]      = acc10[r] + bi0;
        cp[(size_t)(m + 16) * EMB + 16 + l16] = acc11[r] + bi1;
    }
}

// ---------------------------------------------------------------------------
// 3) S[b,a,c] = (sum_s K[b,s,a] * Q[b,s,c]) / sqrt(EMB), then mask -> -1e9
//    M-dim = a, N-dim = c, reduction over s (2048). Wave = 32x32 tile.
// ---------------------------------------------------------------------------
__global__ void s_gemm(const float* __restrict__ Km,           // (B,S,E)
                       const float* __restrict__ Qm,           // (B,S,E)
                       const unsigned char* __restrict__ mask, // (B,E,E) bool
                       float* __restrict__ S)                  // (B,E,E)
{
    const int c0   = blockIdx.x * 32;
    const int a0   = blockIdx.y * 32;
    const int b    = blockIdx.z;
    const int lane = threadIdx.x;
    const int half = lane >> 4;
    const int l16  = lane & 15;

    const float* Kb = Km + (size_t)b * SEQ * EMB;
    const float* Qb = Qm + (size_t)b * SEQ * EMB;
    const float* ka0 = Kb + a0 + l16;        // K^T rows = K columns
    const float* ka1 = ka0 + 16;
    const float* qc0 = Qb + c0 + l16;
    const float* qc1 = qc0 + 16;

    v8f acc00 = {}, acc01 = {}, acc10 = {}, acc11 = {};
#pragma unroll 2
    for (int s = 0; s < SEQ; s += 4) {
        const size_t ks = (size_t)(s + 2 * half) * EMB;
        v2f a0, a1, q0, q1;
        a0.x = ka0[ks];       a0.y = ka0[ks + EMB];
        a1.x = ka1[ks];       a1.y = ka1[ks + EMB];
        q0.x = qc0[ks];       q0.y = qc0[ks + EMB];
        q1.x = qc1[ks];       q1.y = qc1[ks + EMB];
        WMMA_F32(acc00, a0, q0);
        WMMA_F32(acc01, a0, q1);
        WMMA_F32(acc10, a1, q0);
        WMMA_F32(acc11, a1, q1);
    }

    const float scale = 0.044194173824159216f;  // 1/sqrt(512)
    const unsigned char* mb = mask + (size_t)b * EMB * EMB;
    float* sb = S + (size_t)b * EMB * EMB;
#pragma unroll
    for (int r = 0; r < 8; ++r) {
        const int a = a0 + r + 8 * half;
        const int c = c0 + l16;
        float v00 = acc00[r] * scale;
        float v01 = acc01[r] * scale;
        float v10 = acc10[r] * scale;
        float v11 = acc11[r] * scale;
        if (mb[(size_t)a * EMB + c])              v00 = -1e9f;
        if (mb[(size_t)a * EMB + c + 16])         v01 = -1e9f;
        if (mb[(size_t)(a + 16) * EMB + c])       v10 = -1e9f;
        if (mb[(size_t)(a + 16) * EMB + c + 16])  v11 = -1e9f;
        sb[(size_t)a * EMB + c]             = v00;
        sb[(size_t)a * EMB + c + 16]        = v01;
        sb[(size_t)(a + 16) * EMB + c]      = v10;
        sb[(size_t)(a + 16) * EMB + c + 16] = v11;
    }
}

// ---------------------------------------------------------------------------
// 4) Row softmax over last axis: rows of 512, one 256-thread block per row.
// ---------------------------------------------------------------------------
__global__ void softmax_rows(const float* __restrict__ S, float* __restrict__ A)
{
    const int row = blockIdx.x;                 // 0 .. B*EMB-1
    const float* in = S + (size_t)row * EMB;
    float* out = A + (size_t)row * EMB;
    __shared__ float red[256];
    const int t = threadIdx.x;

    const float v0 = in[t];
    const float v1 = in[t + 256];

    red[t] = fmaxf(v0, v1);
    __syncthreads();
    for (int off = 128; off > 0; off >>= 1) {
        if (t < off) red[t] = fmaxf(red[t], red[t + off]);
        __syncthreads();
    }
    const float rmax = red[0];
    __syncthreads();

    const float e0 = __expf(v0 - rmax);
    const float e1 = __expf(v1 - rmax);
    red[t] = e0 + e1;
    __syncthreads();
    for (int off = 128; off > 0; off >>= 1) {
        if (t < off) red[t] += red[t + off];
        __syncthreads();
    }
    const float inv = 1.0f / red[0];
    out[t]       = e0 * inv;
    out[t + 256] = e1 * inv;
}

// ---------------------------------------------------------------------------
// 5) out[b,s,c] = sum_e V[b,s,e] * A[b,e,c].  Wave = 32x32 tile.
// ---------------------------------------------------------------------------
__global__ void out_gemm(const float* __restrict__ V,   // (B,S,E)
                         const float* __restrict__ Am,  // (B,E,E)
                         float* __restrict__ O)         // (B,S,E)
{
    const int c0   = blockIdx.x * 32;
    const int s0   = blockIdx.y * 32;
    const int b    = blockIdx.z;
    const int lane = threadIdx.x;
    const int half = lane >> 4;
    const int l16  = lane & 15;

    const float* v0p = V + ((size_t)b * SEQ + s0 + l16) * EMB + 2 * half;
    const float* v1p = v0p + (size_t)16 * EMB;
    const float* ac0 = Am + (size_t)b * EMB * EMB + c0 + l16;
    const float* ac1 = ac0 + 16;

    v8f acc00 = {}, acc01 = {}, acc10 = {}, acc11 = {};
#pragma unroll 2
    for (int e = 0; e < EMB; e += 4) {
        const v2f a0 = *(const v2f*)(v0p + e);
        const v2f a1 = *(const v2f*)(v1p + e);
        const size_t ke = (size_t)(e + 2 * half) * EMB;
        v2f p0, p1;
        p0.x = ac0[ke];   p0.y = ac0[ke + EMB];
        p1.x = ac1[ke];   p1.y = ac1[ke + EMB];
        WMMA_F32(acc00, a0, p0);
        WMMA_F32(acc01, a0, p1);
        WMMA_F32(acc10, a1, p0);
        WMMA_F32(acc11, a1, p1);
    }

    float* op = O + ((size_t)b * SEQ + s0) * EMB + c0;
#pragma unroll
    for (int r = 0; r < 8; ++r) {
        const int m = r + 8 * half;
        op[(size_t)m * EMB + l16]             = acc00[r];
        op[(size_t)m * EMB + 16 + l16]        = acc01[r];
        op[(size_t)(m + 16) * EMB + l16]      = acc10[r];
        op[(size_t)(m + 16) * EMB + 16 + l16] = acc11[r];
    }
}

// ---------------------------------------------------------------------------
// Launch
// ---------------------------------------------------------------------------
extern "C" void kernel_launch(void* const* d_in, const int* in_sizes, int n_in,
                              void* d_out, int out_size, void* d_ws, size_t ws_size,
                              hipStream_t stream)
{
    const int*           x    = (const int*)d_in[0];
    const int*           z    = (const int*)d_in[1];
    const unsigned char* mask = (const unsigned char*)d_in[2];
    const float*         Wemb = (const float*)d_in[3];
    const float*         Wpos = (const float*)d_in[4];
    const float*         Wq   = (const float*)d_in[5];
    const float*         bq   = (const float*)d_in[6];
    const float*         Wk   = (const float*)d_in[7];
    const float*         bk   = (const float*)d_in[8];
    const float*         Wv   = (const float*)d_in[9];
    const float*         bv   = (const float*)d_in[10];
    float* out = (float*)d_out;

    const size_t nBSE = (size_t)BATCH * SEQ * EMB;   // 4 Mi floats
    const size_t nBEE = (size_t)BATCH * EMB * EMB;   // 1 Mi floats
    float* ws   = (float*)d_ws;
    float* embX = ws;
    float* embZ = embX + nBSE;
    float* Qb   = embZ + nBSE;
    float* Kb   = Qb + nBSE;
    float* Vb   = Kb + nBSE;
    float* Sb   = Vb + nBSE;
    float* Ab   = Sb + nBEE;

    embed_kernel<<<BATCH * SEQ, 256, 0, stream>>>(x, Wemb, Wpos, embX);
    embed_kernel<<<BATCH * SEQ, 256, 0, stream>>>(z, Wemb, Wpos, embZ);

    dim3 gProj(EMB / 32, SEQ / 32, BATCH);
    proj_gemm<<<gProj, 32, 0, stream>>>(embX, Wq, bq, Qb);
    proj_gemm<<<gProj, 32, 0, stream>>>(embZ, Wk, bk, Kb);
    proj_gemm<<<gProj, 32, 0, stream>>>(embZ, Wv, bv, Vb);

    dim3 gS(EMB / 32, EMB / 32, BATCH);
    s_gemm<<<gS, 32, 0, stream>>>(Kb, Qb, mask, Sb);

    softmax_rows<<<BATCH * EMB, 256, 0, stream>>>(Sb, Ab);

    dim3 gO(EMB / 32, SEQ / 32, BATCH);
    out_gemm<<<gO, 32, 0, stream>>>(Vb, Ab, out);
}